// SpikeSEGPreprocessor_79362405695875
// MI455X (gfx1250) — compile-verified
//
#include <hip/hip_runtime.h>
#include <stdint.h>

namespace {
constexpr int   kH = 720, kW = 1280, kC = 2;
constexpr int   kTEvt = 10, kTTot = 20;
constexpr int   kNPix  = kC * kH * kW;     // 1,843,200
constexpr int   kNPix4 = kNPix / 4;        // 460,800 (exactly 1800 * 256)
constexpr int   kStepBlocks = kNPix4 / 256;
constexpr float kDecay  = 0.9048374180359595f;  // float32(exp(-1/10))
constexpr float kAlpha  = 0.1f;
constexpr float kTarget = 0.1f;
constexpr float kThrMin = 0.1f, kThrMax = 10.0f;
}

// d_ws layout (uint32 view):
//   [0] = tmin bits, [1] = tmax bits, [2..21] = per-step spike counts (u32)
//   byte offset 128: mem[kNPix] floats (7.37 MB)

// ---- zero event frames of d_out + mem state, init scalars (every launch) ----
__global__ __launch_bounds__(256) void k_init(float4* __restrict__ out_evt,
                                              float4* __restrict__ mem,
                                              unsigned* __restrict__ ws) {
  const int n_out4  = kTEvt * kNPix4;
  const int tid0    = blockIdx.x * blockDim.x + threadIdx.x;
  const int stride  = gridDim.x * blockDim.x;
  const float4 z = make_float4(0.f, 0.f, 0.f, 0.f);
  for (int i = tid0; i < n_out4; i += stride) out_evt[i] = z;
  for (int i = tid0; i < kNPix4; i += stride) mem[i] = z;
  if (blockIdx.x == 0 && threadIdx.x < 32) {
    if (threadIdx.x == 0) ws[0] = 0x7F800000u;   // +inf bits (t >= 0 => uint order == float order)
    if (threadIdx.x == 1) ws[1] = 0u;            // 0.0f
    if (threadIdx.x >= 2 && threadIdx.x < 2 + kTTot) ws[threadIdx.x] = 0u;
  }
}

// ---- exact t.min / t.max via uint-bit atomics (t is non-negative) ----
__global__ __launch_bounds__(256) void k_minmax(const float* __restrict__ t, int n,
                                                unsigned* __restrict__ ws) {
  float lo = __uint_as_float(0x7F800000u);  // +inf
  float hi = 0.0f;
  const int n4     = n >> 2;
  const int tid0   = blockIdx.x * blockDim.x + threadIdx.x;
  const int stride = gridDim.x * blockDim.x;
  const float4* t4 = (const float4*)t;
  for (int i = tid0; i < n4; i += stride) {
    float4 v = t4[i];                      // global_load_b128
    lo = fminf(lo, fminf(fminf(v.x, v.y), fminf(v.z, v.w)));
    hi = fmaxf(hi, fmaxf(fmaxf(v.x, v.y), fmaxf(v.z, v.w)));
  }
  if (blockIdx.x == 0 && threadIdx.x < (n & 3)) {  // tail
    float v = t[n4 * 4 + threadIdx.x];
    lo = fminf(lo, v);
    hi = fmaxf(hi, v);
  }
  for (int o = 16; o > 0; o >>= 1) {       // wave32 reduction
    lo = fminf(lo, __shfl_down(lo, o, 32));
    hi = fmaxf(hi, __shfl_down(hi, o, 32));
  }
  __shared__ float slo[8], shi[8];
  const int wave = threadIdx.x >> 5, lane = threadIdx.x & 31;
  if (lane == 0) { slo[wave] = lo; shi[wave] = hi; }
  __syncthreads();
  if (threadIdx.x == 0) {
    for (int w = 1; w < 8; ++w) { lo = fminf(lo, slo[w]); hi = fmaxf(hi, shi[w]); }
    atomicMin(&ws[0], __float_as_uint(lo));
    atomicMax(&ws[1], __float_as_uint(hi));
  }
}

// ---- scatter events into first 10 frames of d_out (exact f32 counter adds) ----
__device__ __forceinline__ void scatter_one(int xe, int ye, int pe, float te,
                                            bool valid, float tmin, float denom,
                                            float* __restrict__ out) {
  float tn = valid ? (((te - tmin) / denom) * 9.999999f) : 0.0f;  // T_EVT - 1e-6
  int ti = (int)tn;                        // truncation == astype(int32) for tn >= 0
  ti = min(max(ti, 0), kTEvt - 1);
  int xi = min(max(xe, 0), kW - 1);
  int yi = min(max(ye, 0), kH - 1);
  size_t off = (((size_t)ti * kC + pe) * kH + yi) * kW + xi;
  atomicAdd(&out[off], 1.0f);
}

__global__ __launch_bounds__(256) void k_scatter(const int4* __restrict__ x4,
                                                 const int4* __restrict__ y4,
                                                 const int4* __restrict__ p4,
                                                 const float4* __restrict__ t4, int n,
                                                 const unsigned* __restrict__ ws,
                                                 float* __restrict__ out) {
  const float tmin  = __uint_as_float(ws[0]);
  const float tmax  = __uint_as_float(ws[1]);
  const bool  valid = tmax > tmin;
  const float denom = valid ? (tmax - tmin) : 1.0f;
  const int n4     = n >> 2;
  const int tid0   = blockIdx.x * blockDim.x + threadIdx.x;
  const int stride = gridDim.x * blockDim.x;
  for (int i = tid0; i < n4; i += stride) {
    int4   xv = x4[i];                     // B128 streaming loads
    int4   yv = y4[i];
    int4   pv = p4[i];
    float4 tv = t4[i];
    scatter_one(xv.x, yv.x, pv.x, tv.x, valid, tmin, denom, out);
    scatter_one(xv.y, yv.y, pv.y, tv.y, valid, tmin, denom, out);
    scatter_one(xv.z, yv.z, pv.z, tv.z, valid, tmin, denom, out);
    scatter_one(xv.w, yv.w, pv.w, tv.w, valid, tmin, denom, out);
  }
  if (blockIdx.x == 0 && threadIdx.x < (n & 3)) {  // tail
    int i = n4 * 4 + threadIdx.x;
    scatter_one(((const int*)x4)[i], ((const int*)y4)[i], ((const int*)p4)[i],
                ((const float*)t4)[i], valid, tmin, denom, out);
  }
}

// ---- one LIF step: frame staged via async global->LDS (ASYNCcnt path) ----
__global__ __launch_bounds__(256) void k_step(float* __restrict__ out_frames,
                                              float* __restrict__ mem,
                                              unsigned* __restrict__ ws, int step) {
  __shared__ float4 tile[256];
  const int  i = blockIdx.x * 256 + threadIdx.x;   // exact cover: kNPix4 threads
  const bool has_frame = (step < kTEvt);
  float4* mem4 = (float4*)mem;
  float4* out4 = (float4*)(out_frames + (size_t)step * kNPix);

  // Kick off async frame load (memory -> LDS); overlaps with the mem load and
  // the threshold-recurrence scalar work below.
  if (has_frame) {
    const float4* g = out4 + i;
    unsigned lds_off = (unsigned)(uintptr_t)(&tile[threadIdx.x]);  // low 32b of flat LDS addr
    asm volatile("global_load_async_to_lds_b128 %0, %1, off"
                 :: "v"(lds_off), "v"(g) : "memory");
  }

  float4 m = mem4[i];

  // Replay exact threshold recurrence from finalized integer spike counts.
  const unsigned* counts = ws + 2;
  float thr = 1.0f;
  for (int s = 0; s < step; ++s) {
    float rate = (float)counts[s] / (float)kNPix;
    thr = thr + kAlpha * (rate - kTarget);
    thr = fminf(fmaxf(thr, kThrMin), kThrMax);
  }

  float4 f = make_float4(0.f, 0.f, 0.f, 0.f);
  if (has_frame) {
    asm volatile("s_wait_asynccnt 0x0" ::: "memory");
    f = tile[threadIdx.x];
  }

  m.x = m.x * kDecay + f.x;
  m.y = m.y * kDecay + f.y;
  m.z = m.z * kDecay + f.z;
  m.w = m.w * kDecay + f.w;

  float4 s;
  s.x = (m.x >= thr) ? 1.0f : 0.0f;
  s.y = (m.y >= thr) ? 1.0f : 0.0f;
  s.z = (m.z >= thr) ? 1.0f : 0.0f;
  s.w = (m.w >= thr) ? 1.0f : 0.0f;

  m.x -= s.x * thr;  m.y -= s.y * thr;
  m.z -= s.z * thr;  m.w -= s.w * thr;

  mem4[i] = m;
  out4[i] = s;   // overwrite frame with spikes (frame already consumed via LDS)

  // Exact integer spike count for this step (deterministic across replays).
  int c = (int)s.x + (int)s.y + (int)s.z + (int)s.w;
  for (int o = 16; o > 0; o >>= 1) c += __shfl_down(c, o, 32);
  __shared__ int wsum[8];
  const int wave = threadIdx.x >> 5, lane = threadIdx.x & 31;
  if (lane == 0) wsum[wave] = c;
  __syncthreads();
  if (threadIdx.x == 0) {
    int tot = 0;
    for (int w = 0; w < 8; ++w) tot += wsum[w];
    atomicAdd(&ws[2 + step], (unsigned)tot);
  }
}

extern "C" void kernel_launch(void* const* d_in, const int* in_sizes, int n_in,
                              void* d_out, int out_size, void* d_ws, size_t ws_size,
                              hipStream_t stream) {
  const int*   x = (const int*)d_in[0];
  const int*   y = (const int*)d_in[1];
  const int*   p = (const int*)d_in[2];
  const float* t = (const float*)d_in[3];
  const int    n = in_sizes[0];

  float*    out = (float*)d_out;
  unsigned* ws  = (unsigned*)d_ws;
  float*    mem = (float*)((char*)d_ws + 128);

  k_init<<<4096, 256, 0, stream>>>((float4*)out, (float4*)mem, ws);
  k_minmax<<<1024, 256, 0, stream>>>(t, n, ws);
  k_scatter<<<2048, 256, 0, stream>>>((const int4*)x, (const int4*)y,
                                      (const int4*)p, (const float4*)t, n, ws, out);
  for (int s = 0; s < kTTot; ++s) {
    k_step<<<kStepBlocks, 256, 0, stream>>>(out, mem, ws, s);
  }
}